// Deepseekv3Gate_5196910428666
// MI455X (gfx1250) — compile-verified
//
#include <hip/hip_runtime.h>

typedef __attribute__((ext_vector_type(2))) float v2f;
typedef __attribute__((ext_vector_type(8))) float v8f;

static constexpr int TOKENS   = 8192;
static constexpr int HIDDEN   = 7168;
static constexpr int EXPERTS  = 256;
static constexpr int N_GROUP  = 8;
static constexpr int GSIZE    = EXPERTS / N_GROUP;   // 32
static constexpr int TOPK_GRP = 4;
static constexpr int TOP_K    = 8;
static constexpr float SCALING = 2.5f;

static constexpr int KC      = 32;    // K chunk staged in LDS
static constexpr int ROWS    = 16;    // tokens per block
static constexpr int THREADS = 512;   // 16 wave32 waves: wave w -> experts [16w,16w+16)
static constexpr int APAD    = KC + 4;       // row stride 36 floats: 16B aligned, conflict-free
static constexpr float NEG_INF = -3.402823466e38f;

__global__ __launch_bounds__(THREADS)
void deepseekv3_gate_kernel(const float* __restrict__ x,
                            const float* __restrict__ w,
                            const float* __restrict__ bias,
                            float* __restrict__ out)
{
    __shared__ __align__(16) float Alds[ROWS][APAD];
    __shared__ __align__(16) float Wlds[EXPERTS][APAD];
    __shared__ __align__(16) float Slds[ROWS][EXPERTS + 1];

    const int tid  = threadIdx.x;
    const int lane = tid & 31;
    const int wave = tid >> 5;
    const int tok0 = blockIdx.x * ROWS;
    const int e0   = wave * 16;          // first expert column of this wave's tile
    const int r    = lane & 15;
    const int half = lane >> 4;          // 0: K{0,1}, 1: K{2,3} within each wmma step

    v8f acc = {};

    // ---------------- GEMM: logits tile [16 tokens x 16 experts] over K=7168 ----------------
    for (int kc = 0; kc < HIDDEN; kc += KC) {
        __syncthreads();
        // stage A chunk: 16 rows x 32 cols = 128 float4 (threads 0..127)
        if (tid < 128) {
            const int row = tid >> 3, q = tid & 7;
            const float4 v4 = *(const float4*)(x + (size_t)(tok0 + row) * HIDDEN + kc + q * 4);
            *(float4*)&Alds[row][q * 4] = v4;
        }
        // stage W chunk: 256 rows x 32 cols = 2048 float4 (4 per thread, coalesced)
        #pragma unroll
        for (int i = 0; i < 4; ++i) {
            const int idx = i * THREADS + tid;
            const int row = idx >> 3, q = idx & 7;
            const float4 v4 = *(const float4*)(w + (size_t)row * HIDDEN + kc + q * 4);
            *(float4*)&Wlds[row][q * 4] = v4;
        }
        __syncthreads();
        // 8 fp32 WMMA steps (K=4 each)
        #pragma unroll
        for (int kk = 0; kk < KC; kk += 4) {
            v2f a = *(const v2f*)&Alds[r][kk + 2 * half];          // A: 16x4, M=r
            v2f b = *(const v2f*)&Wlds[e0 + r][kk + 2 * half];     // B: 4x16, N=r (W row-major)
            acc = __builtin_amdgcn_wmma_f32_16x16x4_f32(
                false, a, false, b, (short)0, acc, false, false);
        }
    }

    // scatter C tile (C/D layout: vgpr v -> row v + 8*half, col = r) into score LDS
    #pragma unroll
    for (int vi = 0; vi < 8; ++vi)
        Slds[vi + half * 8][e0 + r] = acc[vi];
    __syncthreads();

    // ---------------- Selection: wave t owns token t (32 lanes x 8 groups) ----------------
    const int t = wave;
    float sig[N_GROUP];    // sigmoid(logit)
    float swbm[N_GROUP];   // sigmoid + bias, later masked / consumed
    float gscore[N_GROUP]; // per-group top-2 sum (identical on all lanes)

    #pragma unroll
    for (int g = 0; g < N_GROUP; ++g) {
        const float logit = Slds[t][g * GSIZE + lane];
        const float s = 1.0f / (1.0f + __expf(-logit));
        sig[g] = s;
        const float v = s + bias[g * GSIZE + lane];
        swbm[g] = v;
        // wave-wide top-2 butterfly reduction over the 32 experts of group g
        float m1 = v, m2 = NEG_INF;
        #pragma unroll
        for (int off = 16; off >= 1; off >>= 1) {
            const float o1 = __shfl_xor(m1, off, 32);
            const float o2 = __shfl_xor(m2, off, 32);
            if (o1 > m1) { m2 = fmaxf(m1, o2); m1 = o1; }
            else         { m2 = fmaxf(m2, o1); }
        }
        gscore[g] = m1 + m2;
    }

    // top-4 groups (redundantly computed per lane; ties -> lowest index, like lax.top_k)
    unsigned keep = 0;
    {
        float gs[N_GROUP];
        #pragma unroll
        for (int g = 0; g < N_GROUP; ++g) gs[g] = gscore[g];
        for (int k = 0; k < TOPK_GRP; ++k) {
            float best = NEG_INF; int bi = 0;
            #pragma unroll
            for (int g = 0; g < N_GROUP; ++g)
                if (gs[g] > best) { best = gs[g]; bi = g; }
            keep |= 1u << bi;
            #pragma unroll
            for (int g = 0; g < N_GROUP; ++g)
                if (g == bi) gs[g] = NEG_INF;
        }
    }
    // mask: reference multiplies by 0/1 mask -> dropped entries become exactly 0.0
    #pragma unroll
    for (int g = 0; g < N_GROUP; ++g)
        if (!((keep >> g) & 1)) swbm[g] = 0.0f;

    // top-8 experts among masked scores; tie-break: lower expert index wins
    unsigned chosen = 0;
    float ssum = 0.0f;
    for (int k = 0; k < TOP_K; ++k) {
        float v = NEG_INF; int e = 1 << 30;
        #pragma unroll
        for (int g = 0; g < N_GROUP; ++g)
            if (swbm[g] > v) { v = swbm[g]; e = g * GSIZE + lane; }
        #pragma unroll
        for (int off = 1; off <= 16; off <<= 1) {
            const float ov = __shfl_xor(v, off, 32);
            const int   oe = __shfl_xor(e, off, 32);
            if (ov > v || (ov == v && oe < e)) { v = ov; e = oe; }
        }
        const int wg = e >> 5, wl = e & 31;
        float cand = 0.0f;
        #pragma unroll
        for (int g = 0; g < N_GROUP; ++g)
            if (g == wg) cand = sig[g];
        ssum += __shfl(cand, wl, 32);          // winner's plain sigmoid score
        if (lane == wl) {
            #pragma unroll
            for (int g = 0; g < N_GROUP; ++g)
                if (g == wg) { swbm[g] = NEG_INF; chosen |= 1u << g; }
        }
    }

    const float scale = SCALING / (ssum + 1e-20f);
    #pragma unroll
    for (int g = 0; g < N_GROUP; ++g) {
        const float o = ((chosen >> g) & 1) ? sig[g] * scale : 0.0f;
        out[(size_t)(tok0 + t) * EXPERTS + g * GSIZE + lane] = o;
    }
}

extern "C" void kernel_launch(void* const* d_in, const int* in_sizes, int n_in,
                              void* d_out, int out_size, void* d_ws, size_t ws_size,
                              hipStream_t stream) {
    const float* x    = (const float*)d_in[0];   // [8192, 7168] fp32
    const float* w    = (const float*)d_in[1];   // [256, 7168]  fp32
    const float* bias = (const float*)d_in[2];   // [256]        fp32
    float* out = (float*)d_out;                  // [8192, 256]  fp32
    (void)in_sizes; (void)n_in; (void)out_size; (void)d_ws; (void)ws_size;

    dim3 grid(TOKENS / ROWS);                    // 512 blocks
    deepseekv3_gate_kernel<<<grid, THREADS, 0, stream>>>(x, w, bias, out);
}